// ParallelMLPWithPE_85066122265223
// MI455X (gfx1250) — compile-verified
//
#include <hip/hip_runtime.h>
#include <math.h>

typedef __attribute__((ext_vector_type(2))) float v2f;
typedef __attribute__((ext_vector_type(8))) float v8f;

#define PI_F 3.14159265358979323846f

// One wave32 per sample. Matvec h = v @ W done as D = A x B with
// V_WMMA_F32_16X16X4_F32: A row 0 = input-vector chunk (lanes 0/16),
// B = raw row-major weight tile loaded straight from global memory into
// the WMMA B layout via per-lane addressing (one base VGPR + immediate
// offsets). D row 0 (VGPR0, lanes 0-15) holds the 16 outputs.
__global__ __launch_bounds__(256) void ParallelMLPWithPE_kernel(
    const float* __restrict__ x,       // [B,2]
    const float* __restrict__ dir,     // [B,3]
    const int*   __restrict__ mlp_idx, // [B]
    const float* __restrict__ w1,      // [M,14,64]
    const float* __restrict__ b1,      // [M,64]
    const float* __restrict__ w2,      // [M,64,16]
    const float* __restrict__ b2,      // [M,16]
    const float* __restrict__ w3,      // [M,31,3]
    const float* __restrict__ b3,      // [M,3]
    float* __restrict__ out,           // [B,3]
    int nB)
{
    __shared__ float lds_h[8][64];   // per-wave hidden1
    __shared__ float lds_g[8][32];   // per-wave concat(h2, dir_pe, pad)

    const int lane = threadIdx.x;          // 0..31
    const int wv   = threadIdx.y;          // 0..7  (one wave each)
    const int s    = blockIdx.x * 8 + wv;  // sample id (uniform per wave)
    if (s >= nB) return;                   // uniform exit: EXEC stays all-1s

    const int   midx = mlp_idx[s];
    const int   hi   = lane >> 4;          // 0: lanes 0-15, 1: lanes 16-31
    const int   col  = lane & 15;
    const bool  head = (col == 0);         // lane 0 or lane 16

    // ---------------- positional encoding of x (14 values, k=14,15 pad=0) ---
    const float x0 = x[2 * s + 0];
    const float x1 = x[2 * s + 1];
    float xpe[16];
    xpe[0] = x0; xpe[1] = x1;
#pragma unroll
    for (int f = 0; f < 3; ++f) {
        const float fr = PI_F * (float)(1 << f);
        xpe[2 + 4 * f + 0] = __sinf(fr * x0);
        xpe[2 + 4 * f + 1] = __sinf(fr * x1);
        xpe[2 + 4 * f + 2] = __cosf(fr * x0);
        xpe[2 + 4 * f + 3] = __cosf(fr * x1);
    }
    xpe[14] = 0.f; xpe[15] = 0.f;

    // A chunks for layer 1: A[0][4c..4c+3] = xpe chunk; all other rows zero.
    v2f A1[4];
#pragma unroll
    for (int c = 0; c < 4; ++c) {
        float a0 = 0.f, a1 = 0.f;
        if (lane == 0)  { a0 = xpe[4 * c + 0]; a1 = xpe[4 * c + 1]; }
        if (lane == 16) { a0 = xpe[4 * c + 2]; a1 = xpe[4 * c + 3]; }
        v2f a; a.x = a0; a.y = a1;
        A1[c] = a;
    }

    // Per-lane base element offsets (constant per lane; loop offsets are
    // compile-time immediates so each load is base-VGPR + IOFFSET).
    const int p1  = 128 * hi + col;                 // layer1 rows 2hi, cols 16t+col
    const int p1x3 = (hi ? 13 : 12) * 64 + col;     // layer1 c=3 clamp (A=0 there)
    const int p1y3 = 13 * 64 + col;
    const int p2  = 32 * hi + col;                  // layer2: (2hi)*16 + col
    const int p3  = 6 * hi + col;                   // layer3: (2hi)*3 + col

    // ---------------- layer 1: h[64] = xpe(14) @ W1(14x64), 4 col-tiles ----
    const float* w1p = w1 + (size_t)midx * (14 * 64);
    const float* b1p = b1 + (size_t)midx * 64;
#pragma unroll
    for (int t = 0; t < 4; ++t) {
        v8f acc = {0.f, 0.f, 0.f, 0.f, 0.f, 0.f, 0.f, 0.f};
#pragma unroll
        for (int c = 0; c < 4; ++c) {
            v2f B;
            if (c < 3) {
                B.x = w1p[p1 + 256 * c + 16 * t];        // row 4c+2hi
                B.y = w1p[p1 + 256 * c + 64 + 16 * t];   // row 4c+2hi+1
            } else {
                B.x = w1p[p1x3 + 16 * t];                // rows clamped <=13
                B.y = w1p[p1y3 + 16 * t];
            }
            acc = __builtin_amdgcn_wmma_f32_16x16x4_f32(
                false, A1[c], false, B, (short)0, acc, false, false);
        }
        float h = acc[0] + b1p[16 * t + col];
        h = fmaxf(h, 0.f);
        if (lane < 16) lds_h[wv][16 * t + lane] = h;   // row0 lives in lanes 0-15
    }
    asm volatile("s_wait_dscnt 0" ::: "memory");

    // ---------------- layer 2: h2[16] = h(64) @ W2(64x16) -------------------
    const float* w2p = w2 + (size_t)midx * (64 * 16);
    const float* hb  = lds_h[wv];
    v8f acc2 = {0.f, 0.f, 0.f, 0.f, 0.f, 0.f, 0.f, 0.f};
#pragma unroll
    for (int c = 0; c < 16; ++c) {
        const float g0 = hb[4 * c + 2 * hi];
        const float g1 = hb[4 * c + 2 * hi + 1];
        v2f A; A.x = head ? g0 : 0.f; A.y = head ? g1 : 0.f;
        v2f B;
        B.x = w2p[p2 + 64 * c];         // row 4c+2hi
        B.y = w2p[p2 + 64 * c + 16];    // row 4c+2hi+1
        acc2 = __builtin_amdgcn_wmma_f32_16x16x4_f32(
            false, A, false, B, (short)0, acc2, false, false);
    }
    float h2 = acc2[0] + b2[(size_t)midx * 16 + col];
    h2 = fmaxf(h2, 0.f);

    // ---------------- build concat vector g[32] = [h2(16), dir_pe(15), 0] ---
    const float d0 = dir[3 * s + 0];
    const float d1 = dir[3 * s + 1];
    const float d2 = dir[3 * s + 2];
    float gval;
    if (lane < 16) {
        gval = h2;
    } else {
        const int e  = lane - 16;                       // 0..15
        const int r3 = e % 3;
        const float dc = (r3 == 0) ? d0 : (r3 == 1) ? d1 : d2;
        float val;
        if (e < 3) {
            val = dc;
        } else {
            const int m     = e - 3;                    // 0..11
            const int f     = m / 6;                    // freq index 0,1
            const int iscos = (m / 3) & 1;
            const float ang = PI_F * (float)(1 << f) * dc;
            val = iscos ? __cosf(ang) : __sinf(ang);
        }
        gval = (e == 15) ? 0.f : val;                   // k=31 pad
    }
    lds_g[wv][lane] = gval;
    asm volatile("s_wait_dscnt 0" ::: "memory");

    // ---------------- layer 3: out[3] = g(31) @ W3(31x3) --------------------
    const float* w3p = w3 + (size_t)midx * 93;
    const float* gb  = lds_g[wv];
    v8f acc3 = {0.f, 0.f, 0.f, 0.f, 0.f, 0.f, 0.f, 0.f};
#pragma unroll
    for (int c = 0; c < 8; ++c) {
        const float t0 = gb[4 * c + 2 * hi];
        const float t1 = gb[4 * c + 2 * hi + 1];
        v2f A; A.x = head ? t0 : 0.f; A.y = head ? t1 : 0.f;
        // rows 4c+2hi(+1); clamp total offset to sample region (cols>=3 and
        // k=31 entries are unread / multiplied by A=0, just need in-bounds).
        int off0 = p3 + 12 * c;      off0 = off0 > 92 ? 92 : off0;
        int off1 = p3 + 12 * c + 3;  off1 = off1 > 92 ? 92 : off1;
        v2f B;
        B.x = w3p[off0];
        B.y = w3p[off1];
        acc3 = __builtin_amdgcn_wmma_f32_16x16x4_f32(
            false, A, false, B, (short)0, acc3, false, false);
    }
    const int bcol = col > 2 ? 2 : col;
    float o = acc3[0] + b3[(size_t)midx * 3 + bcol];
    o = __fdividef(1.f, 1.f + __expf(-o));
    if (lane < 3) out[(size_t)s * 3 + lane] = o;
}

extern "C" void kernel_launch(void* const* d_in, const int* in_sizes, int n_in,
                              void* d_out, int out_size, void* d_ws, size_t ws_size,
                              hipStream_t stream) {
    const float* x    = (const float*)d_in[0];
    const float* dirp = (const float*)d_in[1];
    const int*   idx  = (const int*)  d_in[2];
    const float* w1   = (const float*)d_in[3];
    const float* b1   = (const float*)d_in[4];
    const float* w2   = (const float*)d_in[5];
    const float* b2   = (const float*)d_in[6];
    const float* w3   = (const float*)d_in[7];
    const float* b3   = (const float*)d_in[8];
    float* o = (float*)d_out;
    const int nB = in_sizes[2];                 // number of samples
    dim3 block(32, 8, 1);                       // 8 wave32s per block
    dim3 grid((nB + 7) / 8, 1, 1);
    hipLaunchKernelGGL(ParallelMLPWithPE_kernel, grid, block, 0, stream,
                       x, dirp, idx, w1, b1, w2, b2, w3, b3, o, nB);
}